// MockRenderer_3667902071210
// MI455X (gfx1250) — compile-verified
//
#include <hip/hip_runtime.h>
#include <math.h>

// ---- problem constants (match reference) ----
#define NG      1024
#define HH      256
#define WW      256
#define TILE    16
#define FXc     300.0f
#define FYc     300.0f
#define CXc     128.0f
#define CYc     128.0f
#define NEARc   0.1f
#define FARc    100.0f
#define CULL_R  3.0f
#define T_TH    0.0001f
#define NCHUNK  (NG / 16)

// -0.5 * log2(e): folded into the quad-form coefficients so the WMMA output
// feeds the native v_exp_f32 (exp2) directly with no scaling VALU work.
#define NEG_HALF_LOG2E (-0.72134752044448169f)

typedef __attribute__((ext_vector_type(2))) float v2f;
typedef __attribute__((ext_vector_type(8))) float v8f;

__device__ __forceinline__ float fast_exp2(float x) {
#if __has_builtin(__builtin_amdgcn_exp2f)
    return __builtin_amdgcn_exp2f(x);   // v_exp_f32
#else
    return exp2f(x);
#endif
}

// lane <-> lane^16 exchange. v_permlanex16_b32 with identity lane-selects is a
// single VALU op (no LDS pipe / DScnt wait, unlike ds_bpermute).
__device__ __forceinline__ float swap_half(float x) {
#if __has_builtin(__builtin_amdgcn_permlanex16)
    unsigned u = __float_as_uint(x);
    unsigned r = __builtin_amdgcn_permlanex16(u, u, 0x76543210u, 0xFEDCBA98u, false, false);
    return __uint_as_float(r);
#else
    return __shfl_xor(x, 16);
#endif
}

// 32-bit LDS byte address of a __shared__ object (generic LDS address keeps the
// LDS offset in its low 32 bits on gfx12xx).
__device__ __forceinline__ unsigned int lds_addr(const void* p) {
    return (unsigned int)(unsigned long long)p;
}

// ------------------------------------------------------------------
// Kernel 1: per-gaussian preprocess.
// uCoeff[n][8]: quad-form coefficients (pre-scaled by -0.5*log2e) vs pixel
//               features [px^2, py^2, px*py, px, py, 1, 0, 0]
// uCull[n][4] : (mx, my, (radius*3)^2, a*maskf)
// uSkey[n]    : maskf ? depth : 1e10
// ------------------------------------------------------------------
__global__ void k_preprocess(const float* __restrict__ mean,
                             const float* __restrict__ qvec,
                             const float* __restrict__ log_svec,
                             const float* __restrict__ alpha,
                             const float* __restrict__ c2w,
                             float* __restrict__ uCoeff,
                             float* __restrict__ uCull,
                             float* __restrict__ uSkey) {
    int n = blockIdx.x * blockDim.x + threadIdx.x;
    if (n >= NG) return;

    float R[3][3], t[3];
    #pragma unroll
    for (int r = 0; r < 3; ++r) {
        #pragma unroll
        for (int c = 0; c < 3; ++c) R[r][c] = c2w[r * 4 + c];
        t[r] = c2w[r * 4 + 3];
    }
    float m0 = mean[n * 3 + 0] - t[0];
    float m1 = mean[n * 3 + 1] - t[1];
    float m2 = mean[n * 3 + 2] - t[2];
    float mc[3];
    #pragma unroll
    for (int j = 0; j < 3; ++j) mc[j] = m0 * R[0][j] + m1 * R[1][j] + m2 * R[2][j];

    float depth = mc[2];
    float zc = fmaxf(depth, 1e-6f);
    float iz = 1.0f / zc;
    float x = mc[0], y = mc[1];
    float mx = x * iz, my = y * iz;

    float J[2][3] = {{iz, 0.0f, -x * iz * iz}, {0.0f, iz, -y * iz * iz}};

    float qw = qvec[n * 4 + 0], qx = qvec[n * 4 + 1];
    float qy = qvec[n * 4 + 2], qz = qvec[n * 4 + 3];
    float qn = rsqrtf(qw * qw + qx * qx + qy * qy + qz * qz);
    qw *= qn; qx *= qn; qy *= qn; qz *= qn;
    float Rq[3][3] = {
        {1.0f - 2.0f * (qy * qy + qz * qz), 2.0f * (qx * qy - qw * qz), 2.0f * (qx * qz + qw * qy)},
        {2.0f * (qx * qy + qw * qz), 1.0f - 2.0f * (qx * qx + qz * qz), 2.0f * (qy * qz - qw * qx)},
        {2.0f * (qx * qz - qw * qy), 2.0f * (qy * qz + qw * qx), 1.0f - 2.0f * (qx * qx + qy * qy)}};

    float s0 = expf(log_svec[n * 3 + 0]);
    float s1 = expf(log_svec[n * 3 + 1]);
    float s2 = expf(log_svec[n * 3 + 2]);
    float sv[3] = {s0 * s0, s1 * s1, s2 * s2};

    // cov3d = Rq diag(s^2) Rq^T
    float C3[3][3];
    #pragma unroll
    for (int i = 0; i < 3; ++i)
        #pragma unroll
        for (int k = 0; k < 3; ++k)
            C3[i][k] = Rq[i][0] * sv[0] * Rq[k][0] + Rq[i][1] * sv[1] * Rq[k][1] + Rq[i][2] * sv[2] * Rq[k][2];

    // JW[i][k] = sum_j J[i][j] * R[k][j]
    float JW[2][3];
    #pragma unroll
    for (int i = 0; i < 2; ++i)
        #pragma unroll
        for (int k = 0; k < 3; ++k)
            JW[i][k] = J[i][0] * R[k][0] + J[i][1] * R[k][1] + J[i][2] * R[k][2];

    // cov2 = JW C3 JW^T (symmetric by construction)
    float tmp[2][3];
    #pragma unroll
    for (int i = 0; i < 2; ++i)
        #pragma unroll
        for (int k = 0; k < 3; ++k)
            tmp[i][k] = JW[i][0] * C3[0][k] + JW[i][1] * C3[1][k] + JW[i][2] * C3[2][k];
    float c00 = tmp[0][0] * JW[0][0] + tmp[0][1] * JW[0][1] + tmp[0][2] * JW[0][2];
    float c01 = tmp[0][0] * JW[1][0] + tmp[0][1] * JW[1][1] + tmp[0][2] * JW[1][2];
    float c11 = tmp[1][0] * JW[1][0] + tmp[1][1] * JW[1][1] + tmp[1][2] * JW[1][2];

    float mm = 0.5f * (c00 + c11);
    float det = c00 * c11 - c01 * c01;
    float radius = sqrtf(mm + sqrtf(fmaxf(mm * mm - det, 0.0f)));

    float r3d = fmaxf(s0, fmaxf(s1, s2));
    float marg = r3d * iz;
    const float half_w = (WW * 0.5f) / FXc;
    const float half_h = (HH * 0.5f) / FYc;
    bool maskf = (depth > NEARc) && (depth < FARc) &&
                 (fabsf(mx) < half_w + marg) && (fabsf(my) < half_h + marg);

    float detc = fmaxf(det, 1e-12f);
    float ia = c11 / detc;
    float ib = -c01 / detc;
    float ic = c00 / detc;
    float a = 1.0f / (1.0f + expf(-alpha[n]));

    const float s = NEG_HALF_LOG2E;
    uCoeff[n * 8 + 0] = s * ia;
    uCoeff[n * 8 + 1] = s * ic;
    uCoeff[n * 8 + 2] = s * 2.0f * ib;
    uCoeff[n * 8 + 3] = s * -2.0f * (ia * mx + ib * my);
    uCoeff[n * 8 + 4] = s * -2.0f * (ic * my + ib * mx);
    uCoeff[n * 8 + 5] = s * (ia * mx * mx + ic * my * my + 2.0f * ib * mx * my);
    uCoeff[n * 8 + 6] = 0.0f;
    uCoeff[n * 8 + 7] = 0.0f;

    float rr = radius * CULL_R;
    uCull[n * 4 + 0] = mx;
    uCull[n * 4 + 1] = my;
    uCull[n * 4 + 2] = rr * rr;
    uCull[n * 4 + 3] = maskf ? a : 0.0f;

    uSkey[n] = maskf ? depth : 1e10f;
}

// ------------------------------------------------------------------
// Kernel 2: stable rank sort by depth key (O(N^2), N=1024) + gather.
// ------------------------------------------------------------------
__global__ void k_sortgather(const float* __restrict__ uCoeff,
                             const float* __restrict__ uCull,
                             const float* __restrict__ uSkey,
                             const float* __restrict__ color,
                             float* __restrict__ sCoeff,
                             float* __restrict__ sCull,
                             float* __restrict__ sColor) {
    int n = blockIdx.x * blockDim.x + threadIdx.x;
    if (n >= NG) return;
    float k = uSkey[n];
    int rank = 0;
    for (int m = 0; m < NG; ++m) {
        float km = uSkey[m];
        rank += (km < k) || (km == k && m < n);   // stable argsort rank
    }
    #pragma unroll
    for (int j = 0; j < 8; ++j) sCoeff[rank * 8 + j] = uCoeff[n * 8 + j];
    #pragma unroll
    for (int j = 0; j < 4; ++j) sCull[rank * 4 + j] = uCull[n * 4 + j];
    #pragma unroll
    for (int j = 0; j < 3; ++j) sColor[rank * 3 + j] = color[n * 3 + j];
}

// ------------------------------------------------------------------
// Kernel 3: tile renderer. 1 block = one 16x16 tile, 256 threads (8 wave32).
// All sorted gaussian data staged into LDS once via async-to-LDS (ASYNCcnt).
// quad(gauss,pixel) via two chained V_WMMA_F32_16X16X4_F32 (K=6 pad 8):
//   A (M=gauss x K): lanes 0-15 K={0,1}, lanes 16-31 K={2,3}
//   B (K x N=pixel): vgpr0 rows {K0,K2}, vgpr1 rows {K1,K3}
//   D: pixel = lane%16; gaussians 0-7 lanes<16, 8-15 lanes>=16
// ------------------------------------------------------------------
__global__ __launch_bounds__(256) void k_render(const float* __restrict__ sCoeff,
                                                const float* __restrict__ sCull,
                                                const float* __restrict__ sColor,
                                                float* __restrict__ out) {
    __shared__ __align__(16) float shCoef[NG * 8];   // 32 KB
    __shared__ __align__(16) float shCol[NG * 3];    // 12 KB, rgb interleaved
    __shared__ float shAeff[NG];                     // 4 KB
    __shared__ unsigned int shAct[2];                // chunk-activity bitmask

    const int tid   = threadIdx.x;
    const int tileX = blockIdx.x;
    const int tileY = blockIdx.y;

    if (tid < 2) shAct[tid] = 0u;

    // ---- async staging: global -> LDS, no VGPR round-trip (ASYNCcnt) ----
    {
        unsigned int ldsC = lds_addr(shCoef);
        #pragma unroll
        for (int it = 0; it < 8; ++it) {            // 8192 floats, 16B/thread/iter
            unsigned int off = (unsigned int)(tid * 16 + it * 4096);
            unsigned int dst = ldsC + off;
            asm volatile("global_load_async_to_lds_b128 %0, %1, %2"
                         :: "v"(dst), "v"(off), "s"(sCoeff) : "memory");
        }
        unsigned int ldsK = lds_addr(shCol);
        #pragma unroll
        for (int it = 0; it < 3; ++it) {            // 3072 floats
            unsigned int off = (unsigned int)(tid * 16 + it * 4096);
            unsigned int dst = ldsK + off;
            asm volatile("global_load_async_to_lds_b128 %0, %1, %2"
                         :: "v"(dst), "v"(off), "s"(sColor) : "memory");
        }
    }

    // ---- tile cull + effective alpha (regular loads; needs ALU) ----
    const float psx = 1.0f / FXc, psy = 1.0f / FYc;
    const float tlx = -CXc / FXc, tly = -CYc / FYc;
    const float tx0 = tlx + tileX * (TILE * psx), tx1 = tx0 + TILE * psx;
    const float ty0 = tly + tileY * (TILE * psy), ty1 = ty0 + TILE * psy;

    for (int i = tid; i < NG; i += 256) {
        float mx = sCull[i * 4 + 0], my = sCull[i * 4 + 1];
        float r2 = sCull[i * 4 + 2], ab = sCull[i * 4 + 3];
        float dxt = fmaxf(fmaxf(tx0 - mx, mx - tx1), 0.0f);
        float dyt = fmaxf(fmaxf(ty0 - my, my - ty1), 0.0f);
        shAeff[i] = (dxt * dxt + dyt * dyt <= r2) ? ab : 0.0f;  // cull + maskf folded
    }

    asm volatile("s_wait_asynccnt 0x0" ::: "memory");
    __syncthreads();

    if (tid < NCHUNK) {
        float mxv = 0.0f;
        #pragma unroll
        for (int j = 0; j < 16; ++j) mxv = fmaxf(mxv, shAeff[tid * 16 + j]);
        if (mxv > 0.0f) atomicOr(&shAct[tid >> 5], 1u << (tid & 31));
    }
    __syncthreads();

    const unsigned int act0 = shAct[0];
    const unsigned int act1 = shAct[1];

    const int  lane = tid & 31;
    const int  wave = tid >> 5;
    const int  col  = lane & 15;
    const bool lo   = (lane < 16);
    const int  rA   = wave * 2, rB = wave * 2 + 1;

    const float pxc = tx0 + (col + 0.5f) * psx;
    const float pyA = ty0 + (rA + 0.5f) * psy;
    const float pyB = ty0 + (rB + 0.5f) * psy;

    // B fragments: constant across all chunks.
    v2f bA1, bA2, bB1, bB2;
    bA1.x = lo ? pxc * pxc : pxc * pyA;  // K0 / K2
    bA1.y = lo ? pyA * pyA : pxc;        // K1 / K3
    bA2.x = lo ? pyA : 0.0f;             // K4 / K6
    bA2.y = lo ? 1.0f : 0.0f;            // K5 / K7
    bB1.x = lo ? pxc * pxc : pxc * pyB;
    bB1.y = lo ? pyB * pyB : pxc;
    bB2.x = lo ? pyB : 0.0f;
    bB2.y = lo ? 1.0f : 0.0f;

    float TA = 1.0f, TB = 1.0f;
    float aR = 0.0f, aG = 0.0f, aBc = 0.0f;
    float bR = 0.0f, bG = 0.0f, bBc = 0.0f;

    for (int c = 0; c < NCHUNK; ++c) {
        unsigned int actW = (c < 32) ? act0 : act1;
        if (!((actW >> (c & 31)) & 1u)) continue;                 // uniform across block
        if (!__any((int)(TA > T_TH || TB > T_TH))) break;         // uniform across wave, exact

        const int base = c * 16;
        const int g    = base + col;
        const int off  = lo ? 0 : 2;
        v2f a1, a2;
        a1.x = shCoef[g * 8 + off + 0];
        a1.y = shCoef[g * 8 + off + 1];
        a2.x = shCoef[g * 8 + off + 4];
        a2.y = shCoef[g * 8 + off + 5];

        const int gOff = lo ? 0 : 8;
        float ae[8];
        #pragma unroll
        for (int v = 0; v < 8; ++v) ae[v] = shAeff[base + gOff + v];

        // ---- both pixel rows: D holds (-0.5*log2e)*quad directly ----
        v8f dA = {};
        dA = __builtin_amdgcn_wmma_f32_16x16x4_f32(false, a1, false, bA1, (short)0, dA, false, false);
        dA = __builtin_amdgcn_wmma_f32_16x16x4_f32(false, a2, false, bA2, (short)0, dA, false, false);
        v8f dB = {};
        dB = __builtin_amdgcn_wmma_f32_16x16x4_f32(false, a1, false, bB1, (short)0, dB, false, false);
        dB = __builtin_amdgcn_wmma_f32_16x16x4_f32(false, a2, false, bB2, (short)0, dB, false, false);

        float wA[8], woA[8], wB[8], woB[8];
        #pragma unroll
        for (int v = 0; v < 8; ++v) wA[v] = ae[v] * fast_exp2(dA[v]);
        #pragma unroll
        for (int v = 0; v < 8; ++v) wB[v] = ae[v] * fast_exp2(dB[v]);
        #pragma unroll
        for (int v = 0; v < 8; ++v) woA[v] = swap_half(wA[v]);    // v_permlanex16_b32
        #pragma unroll
        for (int v = 0; v < 8; ++v) woB[v] = swap_half(wB[v]);

        // ---- sequential front-to-back composite, both rows share colors ----
        #pragma unroll
        for (int i = 0; i < 16; ++i) {
            float wa, wb;
            if (i < 8) { wa = lo ? wA[i] : woA[i];         wb = lo ? wB[i] : woB[i]; }
            else       { wa = lo ? woA[i - 8] : wA[i - 8]; wb = lo ? woB[i - 8] : wB[i - 8]; }
            float cr = shCol[3 * (base + i) + 0];
            float cg = shCol[3 * (base + i) + 1];
            float cb = shCol[3 * (base + i) + 2];
            if (TA > T_TH) { float ww = TA * wa; aR += ww * cr; aG += ww * cg; aBc += ww * cb; }
            TA *= (1.0f - wa);
            if (TB > T_TH) { float ww = TB * wb; bR += ww * cr; bG += ww * cg; bBc += ww * cb; }
            TB *= (1.0f - wb);
        }
    }

    // lanes 0-15 hold the (deduplicated) result for their pixel column
    if (lo) {
        const int gx  = tileX * TILE + col;
        const int gyA = tileY * TILE + rA;
        const int gyB = tileY * TILE + rB;
        size_t oA = ((size_t)gyA * WW + gx) * 3;
        size_t oB = ((size_t)gyB * WW + gx) * 3;
        out[oA + 0] = aR; out[oA + 1] = aG; out[oA + 2] = aBc;
        out[oB + 0] = bR; out[oB + 1] = bG; out[oB + 2] = bBc;
    }
}

// ------------------------------------------------------------------
extern "C" void kernel_launch(void* const* d_in, const int* in_sizes, int n_in,
                              void* d_out, int out_size, void* d_ws, size_t ws_size,
                              hipStream_t stream) {
    const float* mean     = (const float*)d_in[0];
    const float* qvec     = (const float*)d_in[1];
    const float* log_svec = (const float*)d_in[2];
    const float* color    = (const float*)d_in[3];
    const float* alpha    = (const float*)d_in[4];
    const float* c2w      = (const float*)d_in[5];
    float* out = (float*)d_out;

    float* w = (float*)d_ws;
    float* uCoeff = w;             // 1024*8
    float* uCull  = w + 8192;      // 1024*4
    float* uSkey  = w + 12288;     // 1024
    float* sCoeff = w + 13312;     // 1024*8
    float* sCull  = w + 21504;     // 1024*4
    float* sColor = w + 25600;     // 1024*3

    k_preprocess<<<dim3(NG / 256), dim3(256), 0, stream>>>(mean, qvec, log_svec, alpha, c2w,
                                                           uCoeff, uCull, uSkey);
    k_sortgather<<<dim3(NG / 256), dim3(256), 0, stream>>>(uCoeff, uCull, uSkey, color,
                                                           sCoeff, sCull, sColor);
    k_render<<<dim3(WW / TILE, HH / TILE), dim3(256), 0, stream>>>(sCoeff, sCull, sColor, out);
}